// UniVarModel_48000554500815
// MI455X (gfx1250) — compile-verified
//
#include <hip/hip_runtime.h>
#include <math.h>

// ---------------------------------------------------------------------------
// GP (sum-of-OU / celerite-style) log-likelihood via Kalman filter, MI455X.
//
// Pipeline (all in d_ws, all launches on `stream`):
//   1. radix sort (4x8-bit LSD) of t -> sorted permutation
//   2. gather: dt[i], y[i], yerr[i]^2 in sorted order
//   3. chunked Kalman filter: 512 chunks of 2048 owned steps, each preceded
//      by 2048 warm-up steps (OU processes forget history exponentially;
//      chunk 0 is exact via a post-warm-up reset). Each wave32 carries TWO
//      chains packed block-diagonally into one 16x16 f32 covariance held in
//      the WMMA C/D register layout; the rank-1 covariance downdate
//      P -= K*Ph^T is one V_WMMA_F32_16X16X4_F32 per timestep (k-slices 0/1
//      keep the two 8x8 blocks independent). Cross-lane reductions use
//      ds_swizzle; phi-row broadcast uses ds_bpermute. Input streams are
//      double-buffered into LDS with the Tensor Data Mover
//      (tensor_load_to_lds + s_wait_tensorcnt). The serial chain uses raw
//      TRANS ops (v_exp_f32 / v_log_f32 / v_rcp_f32) -- the loop is latency
//      bound, and one reciprocal is shared by gain and likelihood terms.
//   4. reduction of per-chunk partial log-likelihoods -> d_out[0] (float)
//
// Workspace requirement: ~30 MB (7N u32 + 256*NB u32 + nchunks doubles).
// ---------------------------------------------------------------------------

typedef float v8f __attribute__((ext_vector_type(8)));
typedef float v2f __attribute__((ext_vector_type(2)));
typedef unsigned int u32;
typedef unsigned int u32x4 __attribute__((ext_vector_type(4)));
typedef unsigned int u32x8 __attribute__((ext_vector_type(8)));

#define TILE      1024
#define CH_OWN    2048
#define WARM      2048        // multiple of BLK
#define BLK       128         // TDM staging tile (elements)
#define LOG2PI_F  1.8378770664093454f
#define LOG2E_F   1.4426950408889634f
#define LN2_F     0.6931471805599453f

// ---- cross-lane helpers (CDNA5 wave32) ------------------------------------
template <int PAT>
__device__ __forceinline__ float swz(float x) {
    return __int_as_float(__builtin_amdgcn_ds_swizzle(__float_as_int(x), PAT));
}
// group-of-32 ds_swizzle encodings: offset[15]=0, [14:10]=xor, [9:5]=or, [4:0]=and
#define SWZ_XOR1   0x041F
#define SWZ_XOR2   0x081F
#define SWZ_XOR4   0x101F
#define SWZ_XOR16  0x401F

// sum within each aligned group of 8 lanes (result replicated in group)
__device__ __forceinline__ float group8_sum(float x) {
    x += swz<SWZ_XOR1>(x);
    x += swz<SWZ_XOR2>(x);
    x += swz<SWZ_XOR4>(x);
    return x;
}

// ---- Tensor Data Mover: 1D tile (BLK f32) global -> LDS --------------------
// D# per CDNA5 ISA. Group0: count=1, lds_addr,
// 57-bit global_addr, type=2. Group1: data_size=4B (code 2), tensor_dim0=td0
// (OOB reads return 0), tensor_dim1=1, tile = BLK x 1, stride=td0.
// All descriptor words forced into SGPRs via readfirstlane (wave-uniform by
// construction). EXEC is ignored by TDM; tracked on TENSORcnt.
__device__ __forceinline__ void tdm_load_1d(u32 ldsoff, const float* gptr, u32 td0) {
    unsigned long long ga = (unsigned long long)(uintptr_t)gptr;
    u32x4 g0;
    g0.x = (u32)__builtin_amdgcn_readfirstlane(1);                       // count=1
    g0.y = (u32)__builtin_amdgcn_readfirstlane((int)ldsoff);             // lds_addr
    g0.z = (u32)__builtin_amdgcn_readfirstlane((int)(u32)ga);            // global_addr[31:0]
    g0.w = (u32)__builtin_amdgcn_readfirstlane(
               (int)((((u32)(ga >> 32)) & 0x01FFFFFFu) | 0x80000000u));  // ga[56:32] | type=2
    u32x8 g1;
    g1[0] = (u32)__builtin_amdgcn_readfirstlane(0x00020000);             // data_size=4B
    g1[1] = (u32)__builtin_amdgcn_readfirstlane((int)((td0 & 0xFFFFu) << 16)); // td0[15:0]
    g1[2] = (u32)__builtin_amdgcn_readfirstlane(
               (int)(((td0 >> 16) & 0xFFFFu) | 0x10000u));               // td0[31:16] | tdim1=1
    g1[3] = (u32)__builtin_amdgcn_readfirstlane((int)(((u32)BLK) << 16)); // tile_dim0=BLK
    g1[4] = (u32)__builtin_amdgcn_readfirstlane(1);                      // tile_dim1=1
    g1[5] = (u32)__builtin_amdgcn_readfirstlane((int)td0);               // dim0 stride
    g1[6] = (u32)__builtin_amdgcn_readfirstlane(0);
    g1[7] = (u32)__builtin_amdgcn_readfirstlane(0);
    asm volatile("tensor_load_to_lds %0, %1" :: "s"(g0), "s"(g1) : "memory");
}

// low 32 bits of a generic pointer to __shared__ == LDS byte offset
__device__ __forceinline__ u32 lds_off(const void* p) {
    return (u32)(unsigned long long)(uintptr_t)p;
}

// ===========================================================================
// 1) radix sort: key prep, per-block histograms, digit-major scan, scatter
// ===========================================================================
__global__ __launch_bounds__(256) void k_makekeys(const float* __restrict__ t,
                                                  u32* __restrict__ keys,
                                                  u32* __restrict__ idx, int n) {
    int i = blockIdx.x * 256 + threadIdx.x;
    if (i < n) {
        u32 u = __float_as_uint(t[i]);
        u ^= (u >> 31) ? 0xFFFFFFFFu : 0x80000000u;  // float -> sortable uint
        keys[i] = u;
        idx[i] = (u32)i;
    }
}

__global__ __launch_bounds__(256) void k_hist(const u32* __restrict__ keys,
                                              u32* __restrict__ hist,
                                              int n, int shift, int nb) {
    __shared__ u32 cnt[256];
    cnt[threadIdx.x] = 0;
    __syncthreads();
    int base = blockIdx.x * TILE;
    #pragma unroll
    for (int k = 0; k < 4; ++k) {
        int i = base + threadIdx.x + k * 256;
        if (i < n) atomicAdd(&cnt[(keys[i] >> shift) & 255u], 1u);
    }
    __syncthreads();
    hist[threadIdx.x * nb + blockIdx.x] = cnt[threadIdx.x];  // digit-major
}

__global__ __launch_bounds__(256) void k_scan(u32* __restrict__ hist, int nb) {
    __shared__ u32 tot[256];
    int d = threadIdx.x;
    u32 s = 0;
    for (int b = 0; b < nb; ++b) s += hist[d * nb + b];
    tot[d] = s;
    __syncthreads();
    if (d == 0) {
        u32 run = 0;
        for (int q = 0; q < 256; ++q) { u32 tmp = tot[q]; tot[q] = run; run += tmp; }
    }
    __syncthreads();
    u32 run = tot[d];
    for (int b = 0; b < nb; ++b) { u32 tmp = hist[d * nb + b]; hist[d * nb + b] = run; run += tmp; }
}

__global__ __launch_bounds__(256) void k_scatter(const u32* __restrict__ keys,
                                                 const u32* __restrict__ idx,
                                                 u32* __restrict__ okeys,
                                                 u32* __restrict__ oidx,
                                                 const u32* __restrict__ hist,
                                                 int n, int shift, int nb) {
    __shared__ u32 sk[TILE];
    __shared__ u32 si[TILE];
    __shared__ u32 sr[TILE];
    __shared__ u32 cnt[256];
    int base = blockIdx.x * TILE;
    #pragma unroll
    for (int k = 0; k < 4; ++k) {
        int i = threadIdx.x + k * 256, g = base + i;
        if (g < n) { sk[i] = keys[g]; si[i] = idx[g]; }
    }
    cnt[threadIdx.x] = 0;
    __syncthreads();
    if (threadIdx.x == 0) {                    // serial stable rank within tile
        int lim = n - base; if (lim > TILE) lim = TILE;
        for (int i = 0; i < lim; ++i) {
            u32 d = (sk[i] >> shift) & 255u;
            sr[i] = cnt[d]++;
        }
    }
    __syncthreads();
    #pragma unroll
    for (int k = 0; k < 4; ++k) {
        int i = threadIdx.x + k * 256, g = base + i;
        if (g < n) {
            u32 d = (sk[i] >> shift) & 255u;
            u32 gp = hist[d * nb + blockIdx.x] + sr[i];
            okeys[gp] = sk[i];
            oidx[gp]  = si[i];
        }
    }
}

// ===========================================================================
// 2) gather sorted streams: dt, y, yerr^2
// ===========================================================================
__global__ __launch_bounds__(256) void k_gather(const u32* __restrict__ idx,
                                                const float* __restrict__ t,
                                                const float* __restrict__ y,
                                                const float* __restrict__ yerr,
                                                float* __restrict__ dts,
                                                float* __restrict__ ys,
                                                float* __restrict__ dsig, int n) {
    int i = blockIdx.x * 256 + threadIdx.x;
    if (i < n) {
        u32 jj = idx[i];
        float ti = t[jj];
        float tp = (i > 0) ? t[idx[i - 1]] : ti;
        dts[i] = ti - tp;                       // dt[0] = 0 as in reference
        ys[i] = y[jj];
        float e = yerr[jj];
        dsig[i] = e * e;
    }
}

// ===========================================================================
// 3) chunked Kalman filter: 2 chains/wave, WMMA downdate, TDM staging
// ===========================================================================
__global__ __launch_bounds__(256) void k_kalman_wmma(const float* __restrict__ dts,
                                                     const float* __restrict__ ys,
                                                     const float* __restrict__ dsig,
                                                     const float* __restrict__ lkp,
                                                     double* __restrict__ partials,
                                                     int n, int nchunks) {
    // per-wave staging: [wave][buf][chain][array][elem]
    __shared__ float stage[8][2][2][3][BLK];

    const int l  = threadIdx.x & 31;
    const int wv = threadIdx.x >> 5;
    const int w  = blockIdx.x * 8 + wv;                  // global wave id
    const int c1 = 2 * w;
    if (c1 >= nchunks) return;                           // wave-uniform exit

    const int col  = l & 15;        // C/D layout column N
    const int half = l >> 4;        // rows M = r (half 0) or r+8 (half 1)
    const int j    = col & 7;       // OU component index
    const int cl   = (col >> 3) & 1;// chain-local id (0/1)
    const int chunk = c1 + cl;

    // kernel params: scale_j = exp(lkp[2j]), sigma_j = exp(lkp[2j+1])
    const float sc     = __expf(lkp[2 * j]);
    const float sg     = __expf(lkp[2 * j + 1]);
    const float pinf   = sg * sg;
    // fold -log2(e)/scale into one per-lane constant: phi = exp2(dt * nls)
    const float nls    = -LOG2E_F * __builtin_amdgcn_rcpf(sc);

    // diagonal masks for element (M=r+8*half, N=col): M==N <=> r==j && half==cl
    float dmask[8];
    #pragma unroll
    for (int r = 0; r < 8; ++r) dmask[r] = (r == j && half == cl) ? 1.0f : 0.0f;

    // state: P in WMMA 16x16 f32 C/D layout (two 8x8 blocks on the diagonal)
    v8f P;
    #pragma unroll
    for (int r = 0; r < 8; ++r) P[r] = dmask[r] * pinf;
    float m = 0.0f;                 // lane-form: lane holds m[col]
    double ll = 0.0;

    const int  sp1  = c1 * CH_OWN - WARM;        // chain-0 stream origin
    const int  sp2  = (c1 + 1) * CH_OWN - WARM;  // chain-1 stream origin
    const bool isC0 = (chunk == 0);
    const int  bidx = half ? 32 : 0;             // bpermute byte base

    const int NBLK = (WARM + CH_OWN) / BLK;

    // issue TDM loads of one BLK-element block for both chains into buf
    auto stage_block = [&](int b, int buf) {
        #pragma unroll
        for (int c = 0; c < 2; ++c) {
            int gbase = (c == 0 ? sp1 : sp2) + b * BLK;
            if (gbase < 0) gbase = 0;            // chunk-0 warm-up: data unused
            u32 td0 = (u32)(n - gbase);          // OOB guard via tensor_dim0
            tdm_load_1d(lds_off(&stage[wv][buf][c][0][0]), dts  + gbase, td0);
            tdm_load_1d(lds_off(&stage[wv][buf][c][1][0]), ys   + gbase, td0);
            tdm_load_1d(lds_off(&stage[wv][buf][c][2][0]), dsig + gbase, td0);
        }
    };

    stage_block(0, 0);                           // prime the pipeline

    for (int b = 0; b < NBLK; ++b) {
        const int buf = b & 1;
        if (b + 1 < NBLK) {
            stage_block(b + 1, buf ^ 1);         // 6 new TDM ops in flight
            // TDM ops from one wave complete in order: <=6 outstanding means
            // block b's 6 tiles have fully landed in LDS.
            __builtin_amdgcn_s_wait_tensorcnt(6);
        } else {
            __builtin_amdgcn_s_wait_tensorcnt(0);
        }
        // keep the LDS reads of the staged tile below the tensorcnt wait
        asm volatile("" ::: "memory");

        for (int q = 0; q < BLK; ++q) {
            const int step = b * BLK + q;
            if (step == WARM) {
                // chunk 0 starts exactly from the stationary prior
                m = isC0 ? 0.0f : m;
                #pragma unroll
                for (int r = 0; r < 8; ++r) P[r] = isC0 ? dmask[r] * pinf : P[r];
            }
            const float dt = stage[wv][buf][cl][0][q];
            const float yv = stage[wv][buf][cl][1][q];
            const float dn = stage[wv][buf][cl][2][q];

            const float phi   = __builtin_amdgcn_exp2f(dt * nls); // v_exp_f32
            const float qdiag = pinf * (1.0f - phi * phi);
            const float mp    = phi * m;

            // P_pred = phi_M * phi_N * P + diag(Pinf*(1-phi^2))
            const int phii = __float_as_int(phi);
            #pragma unroll
            for (int r = 0; r < 8; ++r) {
                float phiM = __int_as_float(
                    __builtin_amdgcn_ds_bpermute(bidx + 4 * r, phii));
                P[r] = phiM * phi * P[r] + dmask[r] * qdiag;
            }

            // Ph = row sums of P_pred; P symmetric => column sums (lane-form)
            float cs = P[0] + P[1] + P[2] + P[3] + P[4] + P[5] + P[6] + P[7];
            cs += swz<SWZ_XOR16>(cs);
            const float Ph = cs;

            // S = sum_j Ph + diag ; v = y - sum_j m_pred (per chain)
            const float S  = group8_sum(Ph) + dn;
            const float v  = yv - group8_sum(mp);
            const float rS = __builtin_amdgcn_rcpf(S);   // one v_rcp_f32,
            const float Kg = Ph * rS;                    // shared by gain and ll
            m = mp + Kg * v;

            // P = P_pred - K*Ph^T via one V_WMMA_F32_16X16X4_F32.
            // k-slice 0: chain 0 (rows/cols 0..7); k-slice 1: chain 1
            // (rows/cols 8..15) -> block-diagonal outer products.
            v2f A, B;
            A[0] = (l < 8)            ? -Kg : 0.0f;
            A[1] = (l >= 8 && l < 16) ? -Kg : 0.0f;
            B[0] = (l < 8)            ?  Ph : 0.0f;
            B[1] = (l >= 8 && l < 16) ?  Ph : 0.0f;
            P = __builtin_amdgcn_wmma_f32_16x16x4_f32(false, A, false, B,
                                                      (short)0, P, false, false);

            if (step >= WARM) {
                const float logS = __builtin_amdgcn_logf(S) * LN2_F; // v_log_f32
                ll += (double)(-0.5f * (LOG2PI_F + logS + v * v * rS));
            }
        }
    }

    if (l == 0) partials[c1] = ll;        // chain 0 (cols 0-7)
    if (l == 8) partials[c1 + 1] = ll;    // chain 1 (cols 8-15)
}

// ===========================================================================
// 4) final reduction
// ===========================================================================
__global__ __launch_bounds__(256) void k_reduce(const double* __restrict__ partials,
                                                int nchunks, float* __restrict__ out) {
    __shared__ double s[256];
    double a = 0.0;
    for (int i = threadIdx.x; i < nchunks; i += 256) a += partials[i];
    s[threadIdx.x] = a;
    __syncthreads();
    for (int off = 128; off > 0; off >>= 1) {
        if (threadIdx.x < off) s[threadIdx.x] += s[threadIdx.x + off];
        __syncthreads();
    }
    if (threadIdx.x == 0) out[0] = (float)s[0];
}

// ===========================================================================
extern "C" void kernel_launch(void* const* d_in, const int* in_sizes, int n_in,
                              void* d_out, int out_size, void* d_ws, size_t ws_size,
                              hipStream_t stream) {
    const float* t    = (const float*)d_in[0];
    const float* y    = (const float*)d_in[1];
    const float* yerr = (const float*)d_in[2];
    const float* lkp  = (const float*)d_in[3];
    const int n = in_sizes[0];

    const int nb = (n + TILE - 1) / TILE;

    // workspace layout (u32 words)
    u32* ws     = (u32*)d_ws;
    u32* keys_a = ws;
    u32* idx_a  = ws + (size_t)n;
    u32* keys_b = ws + (size_t)2 * n;
    u32* idx_b  = ws + (size_t)3 * n;
    u32* hist   = ws + (size_t)4 * n;                         // 256*nb words
    float* dts  = (float*)(ws + (size_t)4 * n + (size_t)256 * nb);
    float* ysr  = dts + n;
    float* dsg  = ysr + n;
    double* partials = (double*)(dsg + n);                    // nchunks doubles

    const int blks = (n + 255) / 256;
    k_makekeys<<<blks, 256, 0, stream>>>(t, keys_a, idx_a, n);

    u32 *ka = keys_a, *ia = idx_a, *kb = keys_b, *ib = idx_b;
    for (int p = 0; p < 4; ++p) {
        const int shift = 8 * p;
        k_hist<<<nb, 256, 0, stream>>>(ka, hist, n, shift, nb);
        k_scan<<<1, 256, 0, stream>>>(hist, nb);
        k_scatter<<<nb, 256, 0, stream>>>(ka, ia, kb, ib, hist, n, shift, nb);
        u32* tk = ka; ka = kb; kb = tk;
        u32* ti = ia; ia = ib; ib = ti;
    }
    // sorted permutation now in `ia`
    k_gather<<<blks, 256, 0, stream>>>(ia, t, y, yerr, dts, ysr, dsg, n);

    const int nchunks = n / CH_OWN;                // N = 1M -> 512 chunks
    const int nwaves  = (nchunks + 1) / 2;         // 2 chains per wave
    const int fblocks = (nwaves + 7) / 8;          // 8 waves per 256-thread block
    k_kalman_wmma<<<fblocks, 256, 0, stream>>>(dts, ysr, dsg, lkp, partials, n, nchunks);

    k_reduce<<<1, 256, 0, stream>>>(partials, nchunks, (float*)d_out);
}